// LocalGrouper_78262894068224
// MI455X (gfx1250) — compile-verified
//
#include <hip/hip_runtime.h>

// ---------------------------------------------------------------------------
// LocalGrouper (PointNet++-style) for MI455X / gfx1250
//   B=16, N=4096, S=1024, K=24, D=64, C=D+3=67, OUTC=2D+3=131
// Phases:
//   1) FPS: 1024 sequential argmax steps; shuffle-based wave argmax +
//      double-buffered LDS partials -> exactly ONE s_barrier per step.
//   2) KNN: V_WMMA_F32_16X16X4_F32 distance tiles fed from LDS-resident
//      batch xyz (48KB staged once per 8-wave block), register top-24.
//      Selection row is read as 4x ds_load_b128 (single wait) into regs.
//   3) Deterministic per-batch std (ddof=1), two-stage reduction.
//   4) Fused affine + concat output (coalesced 206MB store stream).
// ---------------------------------------------------------------------------

#define B_    16
#define N_    4096
#define S_    1024
#define K_    24
#define D_    64
#define C_    67      // D+3
#define OUTC_ 131     // 2D+3

typedef __attribute__((ext_vector_type(2))) float v2f;
typedef __attribute__((ext_vector_type(4))) float v4f;
typedef __attribute__((ext_vector_type(8))) float v8f;

// ---------------------------------------------------------------------------
// Phase 1: farthest point sampling. One block per batch, 256 threads (8 waves),
// 16 points per thread in registers. Per step: intra-wave __shfl_xor argmax
// (no barrier), lane0 writes per-wave partial into parity buffer s&1, one
// __syncthreads, then all threads redundantly reduce the 8 partials.
// First-occurrence (lowest global index) tie-break matches jnp.argmax.
// ---------------------------------------------------------------------------
__global__ __launch_bounds__(256) void fps_kernel(const float* __restrict__ xyz,
                                                  int* __restrict__ fps_idx,
                                                  float* __restrict__ new_xyz) {
  const int b    = blockIdx.x;
  const int tid  = threadIdx.x;
  const int wid  = tid >> 5;
  const int lane = tid & 31;
  const float* bx = xyz + (size_t)b * N_ * 3;

  float px[16], py[16], pz[16], dd[16];
#pragma unroll
  for (int i = 0; i < 16; ++i) {
    const int p = tid + 256 * i;           // strided ownership (coalesced)
    px[i] = bx[p * 3 + 0];
    py[i] = bx[p * 3 + 1];
    pz[i] = bx[p * 3 + 2];
    dd[i] = 1e10f;
  }

  __shared__ float swv[2][8];
  __shared__ int   swi[2][8];

  int last = 0;
  for (int s = 0; s < S_; ++s) {
    if (tid == 0) fps_idx[b * S_ + s] = last;

    // uniform-address load of current farthest point (scalar-load friendly)
    const float lx = bx[last * 3 + 0];
    const float ly = bx[last * 3 + 1];
    const float lz = bx[last * 3 + 2];
    if (tid == 0) {
      new_xyz[(b * S_ + s) * 3 + 0] = lx;
      new_xyz[(b * S_ + s) * 3 + 1] = ly;
      new_xyz[(b * S_ + s) * 3 + 2] = lz;
    }

    float bv = -1.0f;
    int   bi = 0;
#pragma unroll
    for (int i = 0; i < 16; ++i) {
      const float dx = px[i] - lx, dy = py[i] - ly, dz = pz[i] - lz;
      const float d = dx * dx + dy * dy + dz * dz;
      dd[i] = fminf(dd[i], d);
      if (dd[i] > bv) { bv = dd[i]; bi = tid + 256 * i; }
    }

    // intra-wave butterfly argmax (wave32), tie -> lower global index
#pragma unroll
    for (int off = 16; off > 0; off >>= 1) {
      const float ov = __shfl_xor(bv, off, 32);
      const int   oi = __shfl_xor(bi, off, 32);
      if (ov > bv || (ov == bv && oi < bi)) { bv = ov; bi = oi; }
    }

    const int buf = s & 1;
    if (lane == 0) { swv[buf][wid] = bv; swi[buf][wid] = bi; }
    __syncthreads();

    // every thread reduces the 8 per-wave partials (no second barrier;
    // parity double-buffer protects against step s+1 overwrites)
    float fv = swv[buf][0];
    int   fi = swi[buf][0];
#pragma unroll
    for (int j = 1; j < 8; ++j) {
      const float ov = swv[buf][j];
      const int   oi = swi[buf][j];
      if (ov > fv || (ov == fv && oi < fi)) { fv = ov; fi = oi; }
    }
    last = fi;
  }
}

// ---------------------------------------------------------------------------
// Phase 2: brute-force KNN. 256-thread block = 8 waves = 8 anchor tiles,
// sharing one LDS copy of the batch's xyz (48KB). Per 16-point chunk, one
// v_wmma_f32_16x16x4_f32 gives the 16x16 dot tile (A: lanes 0-15 hold {x,y}
// of anchor m=lane, lanes 16-31 hold {z,0}; B analogous for points, per the
// ISA 16x4 f32 operand layout). dist = |a|^2 + |p|^2 - 2*dot goes through a
// wave-private LDS tile; lanes 0-15 keep a register-resident top-24.
// Only ONE block-wide barrier in the whole kernel; per-chunk ordering relies
// on same-wave in-order LDS + __builtin_amdgcn_wave_barrier().
// ---------------------------------------------------------------------------
__global__ __launch_bounds__(256) void knn_kernel(const float* __restrict__ xyz,
                                                  const int* __restrict__ fps_idx,
                                                  int* __restrict__ knn_idx) {
  const int b    = blockIdx.x >> 3;             // 8 blocks per batch
  const int grp  = blockIdx.x & 7;
  const int w    = threadIdx.x >> 5;            // wave id 0..7
  const int lane = threadIdx.x & 31;
  const int hn   = lane & 15;

  const float* bx = xyz + (size_t)b * N_ * 3;

  __shared__ float sxyz[N_ * 3];                // 48KB: whole batch xyz
  __shared__ float tile_d[8][256];              // per-wave distance tile
  __shared__ float s2s[8][16];                  // per-wave |anchor|^2

  // cooperative stage of batch xyz into LDS (12288 floats / 256 threads)
#pragma unroll
  for (int i = 0; i < 48; ++i) {
    const int idx = threadIdx.x + 256 * i;
    sxyz[idx] = bx[idx];
  }
  __syncthreads();                              // the only block barrier

  const int tile   = grp * 8 + w;               // 0..63
  const int anchor = tile * 16 + hn;
  const int aidx   = fps_idx[b * S_ + anchor];
  const float ax = sxyz[aidx * 3 + 0];
  const float ay = sxyz[aidx * 3 + 1];
  const float az = sxyz[aidx * 3 + 2];

  if (lane < 16) s2s[w][lane] = ax * ax + ay * ay + az * az;
  __builtin_amdgcn_wave_barrier();

  v2f A;
  A.x = (lane < 16) ? ax : az;
  A.y = (lane < 16) ? ay : 0.0f;

  // hoist the 8 row |a|^2 values into registers (loop-invariant)
  const int mbase = (lane < 16) ? 0 : 8;
  float s2r[8];
#pragma unroll
  for (int r = 0; r < 8; ++r) s2r[r] = s2s[w][mbase + r];

  // register-resident top-K (replace-worst)
  float hv[K_];
  int   hi[K_];
#pragma unroll
  for (int j = 0; j < K_; ++j) { hv[j] = 3.0e38f; hi[j] = 0; }
  float worst = 3.0e38f;
  int   wslot = 0;

  for (int cBase = 0; cBase < N_; cBase += 16) {
    const int n = cBase + hn;
    const float pxx = sxyz[n * 3 + 0];
    const float pyy = sxyz[n * 3 + 1];
    const float pzz = sxyz[n * 3 + 2];
    const float d2  = pxx * pxx + pyy * pyy + pzz * pzz;

    v2f Bv;
    Bv.x = (lane < 16) ? pxx : pzz;
    Bv.y = (lane < 16) ? pyy : 0.0f;

    v8f Cv = {};
    Cv = __builtin_amdgcn_wmma_f32_16x16x4_f32(
        /*neg_a=*/false, A, /*neg_b=*/false, Bv,
        /*c_mod=*/(short)0, Cv, /*reuse_a=*/false, /*reuse_b=*/false);

#pragma unroll
    for (int r = 0; r < 8; ++r) {
      const float dist = s2r[r] + d2 - 2.0f * Cv[r];
      tile_d[w][(mbase + r) * 16 + hn] = dist;
    }
    __builtin_amdgcn_wave_barrier();            // same-wave DS is in-order

    if (lane < 16) {
      // read the whole 64B row with 4 wide LDS loads -> single wait,
      // then scan candidates entirely from registers
      const v4f* rowp = (const v4f*)&tile_d[w][lane * 16];
      const v4f r0 = rowp[0];
      const v4f r1 = rowp[1];
      const v4f r2 = rowp[2];
      const v4f r3 = rowp[3];
      float row[16];
      row[0]  = r0.x; row[1]  = r0.y; row[2]  = r0.z; row[3]  = r0.w;
      row[4]  = r1.x; row[5]  = r1.y; row[6]  = r1.z; row[7]  = r1.w;
      row[8]  = r2.x; row[9]  = r2.y; row[10] = r2.z; row[11] = r2.w;
      row[12] = r3.x; row[13] = r3.y; row[14] = r3.z; row[15] = r3.w;
#pragma unroll
      for (int j = 0; j < 16; ++j) {
        const float d   = row[j];
        const int   idx = cBase + j;
        if (d < worst) {
#pragma unroll
          for (int t = 0; t < K_; ++t)
            if (t == wslot) { hv[t] = d; hi[t] = idx; }
          worst = -3.0e38f;
#pragma unroll
          for (int t = 0; t < K_; ++t)
            if (hv[t] > worst) { worst = hv[t]; wslot = t; }
        }
      }
    }
    __builtin_amdgcn_wave_barrier();
  }

  // emit sorted ascending (matches top_k of -sq), tie -> lower index
  if (lane < 16) {
    int* outp = knn_idx + ((size_t)(b * S_ + anchor)) * K_;
#pragma unroll 1
    for (int k = 0; k < K_; ++k) {
      float best = 3.0e38f;
      int bidx = 0x40000000, bslot = 0;
#pragma unroll
      for (int t = 0; t < K_; ++t) {
        if (hv[t] < best || (hv[t] == best && hi[t] < bidx)) {
          best = hv[t]; bidx = hi[t]; bslot = t;
        }
      }
      outp[k] = bidx;
#pragma unroll
      for (int t = 0; t < K_; ++t)
        if (t == bslot) hv[t] = 3.0e38f;
    }
  }
}

// ---------------------------------------------------------------------------
// Phase 3: per-batch sum / sumsq of diff (deterministic: per-block partials).
// One thread per (s,k): 67 diff channels. grid = (96, B).
// ---------------------------------------------------------------------------
__global__ __launch_bounds__(256) void stats_kernel(const float* __restrict__ xyz,
                                                    const float* __restrict__ points,
                                                    const int* __restrict__ fps_idx,
                                                    const int* __restrict__ knn_idx,
                                                    float* __restrict__ partials) {
  const int b = blockIdx.y;
  const int t = blockIdx.x * 256 + threadIdx.x;  // S_*K_ = 24576 = 96*256 exactly

  const int s = t / K_;
  const int k = t % K_;
  const int gi = knn_idx[(b * S_ + s) * K_ + k];
  const int fi = fps_idx[b * S_ + s];

  const float* gp = points + ((size_t)b * N_ + gi) * D_;
  const float* mp = points + ((size_t)b * N_ + fi) * D_;
  float sum = 0.0f, ss = 0.0f;
#pragma unroll 8
  for (int c = 0; c < D_; ++c) {
    const float d = gp[c] - mp[c];
    sum += d; ss += d * d;
  }
  const float* gx = xyz + ((size_t)b * N_ + gi) * 3;
  const float* mx = xyz + ((size_t)b * N_ + fi) * 3;
#pragma unroll
  for (int c = 0; c < 3; ++c) {
    const float d = gx[c] - mx[c];
    sum += d; ss += d * d;
  }

  __shared__ float r1[256], r2[256];
  r1[threadIdx.x] = sum;
  r2[threadIdx.x] = ss;
  __syncthreads();
  for (int off = 128; off > 0; off >>= 1) {
    if (threadIdx.x < off) {
      r1[threadIdx.x] += r1[threadIdx.x + off];
      r2[threadIdx.x] += r2[threadIdx.x + off];
    }
    __syncthreads();
  }
  if (threadIdx.x == 0) {
    partials[(b * gridDim.x + blockIdx.x) * 2 + 0] = r1[0];
    partials[(b * gridDim.x + blockIdx.x) * 2 + 1] = r2[0];
  }
}

// Fixed-order reduce of 96 partials per batch -> 1/(std+1e-5) (ddof=1).
__global__ void reduce_kernel(const float* __restrict__ partials,
                              float* __restrict__ invStd) {
  const int b = threadIdx.x;
  if (b >= B_) return;
  double s = 0.0, ss = 0.0;
  for (int p = 0; p < 96; ++p) {
    s  += (double)partials[(b * 96 + p) * 2 + 0];
    ss += (double)partials[(b * 96 + p) * 2 + 1];
  }
  const double n   = (double)S_ * K_ * C_;
  double var = (ss - s * s / n) / (n - 1.0);
  if (var < 0.0) var = 0.0;
  const double sd = sqrt(var);
  invStd[b] = (float)(1.0 / (sd + 1e-5));
}

// ---------------------------------------------------------------------------
// Phase 4: write [B,S,K,131]. Consecutive threads -> consecutive channels
// (coalesced 206MB store stream); gathered reads hit L2 (points = 16.8MB).
// ---------------------------------------------------------------------------
__global__ __launch_bounds__(256) void out_kernel(const float* __restrict__ xyz,
                                                  const float* __restrict__ points,
                                                  const float* __restrict__ alpha,
                                                  const float* __restrict__ beta,
                                                  const int* __restrict__ fps_idx,
                                                  const int* __restrict__ knn_idx,
                                                  const float* __restrict__ invStd,
                                                  float* __restrict__ outp) {
  const size_t e = (size_t)blockIdx.x * 256 + threadIdx.x;
  const size_t total = (size_t)B_ * S_ * K_ * OUTC_;
  if (e >= total) return;

  const int ch     = (int)(e % OUTC_);
  const size_t bsk = e / OUTC_;
  const int k      = (int)(bsk % K_);
  const size_t bs  = bsk / K_;
  const int s      = (int)(bs % S_);
  const int b      = (int)(bs / S_);

  const int fi = fps_idx[b * S_ + s];
  float v;
  if (ch < C_) {
    const int gi = knn_idx[(b * S_ + s) * K_ + k];
    float g, m;
    if (ch < D_) {
      g = points[((size_t)b * N_ + gi) * D_ + ch];
      m = points[((size_t)b * N_ + fi) * D_ + ch];
    } else {
      g = xyz[((size_t)b * N_ + gi) * 3 + (ch - D_)];
      m = xyz[((size_t)b * N_ + fi) * 3 + (ch - D_)];
    }
    v = alpha[ch] * ((g - m) * invStd[b]) + beta[ch];
  } else {
    v = points[((size_t)b * N_ + fi) * D_ + (ch - C_)];
  }
  outp[e] = v;
}

// ---------------------------------------------------------------------------
extern "C" void kernel_launch(void* const* d_in, const int* in_sizes, int n_in,
                              void* d_out, int out_size, void* d_ws, size_t ws_size,
                              hipStream_t stream) {
  (void)in_sizes; (void)n_in; (void)out_size; (void)ws_size;
  const float* xyz    = (const float*)d_in[0];
  const float* points = (const float*)d_in[1];
  const float* alpha  = (const float*)d_in[2];
  const float* beta   = (const float*)d_in[3];
  float* out = (float*)d_out;
  char*  ws  = (char*)d_ws;

  int*   fps_idx  = (int*)ws;                                  // 16384 ints
  int*   knn_idx  = (int*)(ws + 65536);                        // 393216 ints
  float* partials = (float*)(ws + 65536 + 393216 * 4);         // 16*96*2 floats
  float* invStd   = (float*)(ws + 65536 + 393216 * 4 + 16 * 96 * 2 * 4);

  float* new_xyz = out;                                        // [B,S,3]
  float* body    = out + (size_t)B_ * S_ * 3;                  // [B,S,K,131]

  fps_kernel<<<B_, 256, 0, stream>>>(xyz, fps_idx, new_xyz);
  knn_kernel<<<B_ * 8, 256, 0, stream>>>(xyz, fps_idx, knn_idx);
  dim3 sg(96, B_);
  stats_kernel<<<sg, 256, 0, stream>>>(xyz, points, fps_idx, knn_idx, partials);
  reduce_kernel<<<1, 32, 0, stream>>>(partials, invStd);

  const size_t total = (size_t)B_ * S_ * K_ * OUTC_;
  const int blocks = (int)((total + 255) / 256);
  out_kernel<<<blocks, 256, 0, stream>>>(xyz, points, alpha, beta,
                                         fps_idx, knn_idx, invStd, body);
}